// HOTModule_6193342841260
// MI455X (gfx1250) — compile-verified
//
#include <hip/hip_runtime.h>

// CDNA5 / gfx1250 implementation of DPC-KNN token clustering.
// Dominant cost is reading x (285 MB -> ~12us @ 23.3 TB/s); the Gram matrix
// (8.6 GFLOP) runs on V_WMMA_F32_16X16X4_F32 in full fp32 since compute is
// far from the bottleneck and the reference's ranking depends on 1e-6 scale
// perturbations. dist (32MB) + x_knn (16MB) stay L2-resident (192MB L2).
// The Gram kernel register-blocks 4x4 WMMA tiles per wave (0.5 operand loads
// per wmma). Tile base addresses are folded into 24-bit instruction offsets
// (one A pointer + one B pointer) and the kernel is __launch_bounds__(128,1)
// so the 16 f32 accumulators (128 VGPRs) live in registers without spilling.

typedef __attribute__((ext_vector_type(2))) float v2f;
typedef __attribute__((ext_vector_type(8))) float v8f;

#define B_   8
#define F_   1024
#define N_   17
#define C_   512
#define TOK  81
#define FT   (F_ / 16)          // 64 16-wide tiles per frame dimension
#define BT   4                  // 4x4 register-blocked tiles per wave
#define FTB  (FT / BT)          // 16 tile-blocks per frame dimension

// ---------------------------------------------------------------------------
// K1: x_knn[b,f,c] = mean_n x[b,f,n,c];  sq[b,f] = sum_c x_knn^2
//     also zero the per-batch dist-max atomics (blocks 0..7).
// ---------------------------------------------------------------------------
__global__ void k_meanpool(const float* __restrict__ x,
                           float* __restrict__ xknn,
                           float* __restrict__ sq,
                           unsigned* __restrict__ dmax) {
    const int row = blockIdx.x;              // b*F_ + f
    const int tid = threadIdx.x;             // 128 threads, 4 channels each
    if (row < B_ && tid == 0) dmax[row] = 0u;

    const float4* __restrict__ x4 =
        (const float4*)(x + (size_t)row * N_ * C_);
    float4 acc = make_float4(0.f, 0.f, 0.f, 0.f);
    #pragma unroll
    for (int n = 0; n < N_; ++n) {
        float4 v = x4[n * (C_ / 4) + tid];
        acc.x += v.x; acc.y += v.y; acc.z += v.z; acc.w += v.w;
    }
    const float inv = 1.0f / (float)N_;
    acc.x *= inv; acc.y *= inv; acc.z *= inv; acc.w *= inv;
    ((float4*)(xknn + (size_t)row * C_))[tid] = acc;

    __shared__ float red[128];
    red[tid] = acc.x * acc.x + acc.y * acc.y + acc.z * acc.z + acc.w * acc.w;
    __syncthreads();
    for (int s = 64; s > 0; s >>= 1) {
        if (tid < s) red[tid] += red[tid + s];
        __syncthreads();
    }
    if (tid == 0) sq[row] = red[0];
}

// ---------------------------------------------------------------------------
// K2: per-batch Gram via WMMA fp32 16x16x4, register-blocked 4x4 tiles/wave.
//     dist = sqrt(max(sq_i + sq_j - 2*g, 0)) / sqrt(C).
//     A-fragment layout (fp32 16x4): lanes 0-15 hold K=k0,k0+1; lanes 16-31
//     hold K=k0+2,k0+3 (row = lane&15). B mirrored; dist is symmetric so the
//     result is robust to the A/B transposition ambiguity.
//     Tile q is at constant byte offset q*16*C_*4 = q*32768 from tile 0 ->
//     folded into the load's 24-bit immediate offset (2 pointers total).
// ---------------------------------------------------------------------------
__global__ void __launch_bounds__(128, 1)
k_gram_dist(const float* __restrict__ xknn,
            const float* __restrict__ sq,
            float* __restrict__ dist,
            unsigned* __restrict__ dmax) {
    const int wave = blockIdx.x * 4 + (threadIdx.x >> 5);
    const int lane = threadIdx.x & 31;
    const int l16  = lane & 15;
    const int hi   = lane >> 4;              // 0: K=k0..k0+1, 1: K=k0+2..k0+3

    const int b   = wave / (FTB * FTB);
    const int rem = wave % (FTB * FTB);
    const int ti4 = rem / FTB;               // 4-tile row block
    const int tj4 = rem % FTB;               // 4-tile col block

    const float* __restrict__ Xb = xknn + (size_t)b * F_ * C_;
    // single base pointer per operand; per-tile offsets are immediates
    const float* __restrict__ abase =
        Xb + (size_t)(ti4 * BT * 16 + l16) * C_ + hi * 2;
    const float* __restrict__ bbase =
        Xb + (size_t)(tj4 * BT * 16 + l16) * C_ + hi * 2;

    v8f acc[BT][BT] = {};
    for (int k = 0; k < C_; k += 4) {
        v2f a[BT], bb[BT];
        #pragma unroll
        for (int q = 0; q < BT; ++q) {
            a[q]  = *(const v2f*)(abase + k + q * (16 * C_));
            bb[q] = *(const v2f*)(bbase + k + q * (16 * C_));
        }
        #pragma unroll
        for (int qi = 0; qi < BT; ++qi)
            #pragma unroll
            for (int qj = 0; qj < BT; ++qj)
                // 8 args: (neg_a, A, neg_b, B, c_mod, C, reuse_a, reuse_b)
                acc[qi][qj] = __builtin_amdgcn_wmma_f32_16x16x4_f32(
                    false, a[qi], false, bb[qj], (short)0, acc[qi][qj],
                    false, false);
    }

    const float inv_sqrtC = 0.044194173824159216f;   // 1/sqrt(512)
    float sqj[BT];
    #pragma unroll
    for (int qj = 0; qj < BT; ++qj)
        sqj[qj] = sq[b * F_ + (tj4 * BT + qj) * 16 + l16];

    float lmax = 0.f;
    #pragma unroll
    for (int qi = 0; qi < BT; ++qi) {
        #pragma unroll
        for (int v = 0; v < 8; ++v) {
            const int M    = v + hi * 8;             // C/D VGPR layout
            const int irow = (ti4 * BT + qi) * 16 + M;
            const float sqi = sq[b * F_ + irow];
            float* __restrict__ drow =
                dist + ((size_t)b * F_ + irow) * F_ + tj4 * BT * 16 + l16;
            #pragma unroll
            for (int qj = 0; qj < BT; ++qj) {
                float d2 = sqi + sqj[qj] - 2.0f * acc[qi][qj][v];
                float d  = sqrtf(fmaxf(d2, 0.0f)) * inv_sqrtC;
                drow[qj * 16] = d;
                lmax = fmaxf(lmax, d);
            }
        }
    }
    // dist >= 0 so uint bit order == float order
    atomicMax(&dmax[b], __float_as_uint(lmax));
}

// ---------------------------------------------------------------------------
// K3: density[b,f] = exp(-(m1^2+m2^2)/2) + noise*1e-6
//     (m1,m2 = two smallest entries of the dist row, self included)
// ---------------------------------------------------------------------------
__global__ void k_density(const float* __restrict__ dist,
                          const float* __restrict__ noise,
                          float* __restrict__ density) {
    const int row = blockIdx.x;              // b*F_ + f
    const int tid = threadIdx.x;             // 256 threads
    const float* __restrict__ r = dist + (size_t)row * F_;

    float m1 = 3.4e38f, m2 = 3.4e38f;
    for (int j = tid; j < F_; j += 256) {
        float v = r[j];
        if (v < m1) { m2 = m1; m1 = v; }
        else if (v < m2) { m2 = v; }
    }
    __shared__ float s1[256], s2[256];
    s1[tid] = m1; s2[tid] = m2;
    __syncthreads();
    for (int s = 128; s > 0; s >>= 1) {
        if (tid < s) {
            float a1 = s1[tid], a2 = s2[tid];
            float b1 = s1[tid + s], b2 = s2[tid + s];
            float lo = fminf(a1, b1);
            float hi = fminf(fmaxf(a1, b1), fminf(a2, b2));
            s1[tid] = lo; s2[tid] = hi;
        }
        __syncthreads();
    }
    if (tid == 0) {
        float mean = 0.5f * (s1[0] * s1[0] + s2[0] * s2[0]);
        density[row] = expf(-mean) + noise[row] * 1e-6f;
    }
}

// ---------------------------------------------------------------------------
// K4: score[b,i] = density[b,i] * min_j (density[b,j] > density[b,i] ?
//                                        dist[b,i,j] : dist_max[b])
// ---------------------------------------------------------------------------
__global__ void k_score(const float* __restrict__ dist,
                        const float* __restrict__ density,
                        const unsigned* __restrict__ dmax,
                        float* __restrict__ score) {
    const int row = blockIdx.x;              // b*F_ + i
    const int b   = row / F_;
    const int tid = threadIdx.x;             // 256 threads
    const float di   = density[row];
    const float dm   = __uint_as_float(dmax[b]);
    const float* __restrict__ r = dist + (size_t)row * F_;
    const float* __restrict__ db = density + (size_t)b * F_;

    float mn = dm;
    for (int j = tid; j < F_; j += 256)
        if (db[j] > di) mn = fminf(mn, r[j]);

    __shared__ float sm[256];
    sm[tid] = mn;
    __syncthreads();
    for (int s = 128; s > 0; s >>= 1) {
        if (tid < s) sm[tid] = fminf(sm[tid], sm[tid + s]);
        __syncthreads();
    }
    if (tid == 0) score[row] = sm[0] * di;
}

// ---------------------------------------------------------------------------
// K5: per batch: top-81 by score (stable: ties -> lower index), sort indices.
// ---------------------------------------------------------------------------
__global__ void k_topk(const float* __restrict__ score,
                       int* __restrict__ index) {
    const int b   = blockIdx.x;
    const int tid = threadIdx.x;             // 256 threads
    __shared__ float sc[F_];
    __shared__ float bv[256];
    __shared__ int   bi[256];
    __shared__ int   chosen[TOK];

    for (int j = tid; j < F_; j += 256) sc[j] = score[b * F_ + j];
    __syncthreads();

    for (int t = 0; t < TOK; ++t) {
        float val = -3.4e38f; int idx = F_;
        for (int j = tid; j < F_; j += 256) {
            float v = sc[j];
            if (v > val || (v == val && j < idx)) { val = v; idx = j; }
        }
        bv[tid] = val; bi[tid] = idx;
        __syncthreads();
        for (int s = 128; s > 0; s >>= 1) {
            if (tid < s) {
                float ov = bv[tid + s]; int oi = bi[tid + s];
                if (ov > bv[tid] || (ov == bv[tid] && oi < bi[tid])) {
                    bv[tid] = ov; bi[tid] = oi;
                }
            }
            __syncthreads();
        }
        if (tid == 0) { chosen[t] = bi[0]; sc[bi[0]] = -3.4e38f; }
        __syncthreads();
    }
    if (tid == 0) {                          // insertion sort, 81 elems
        for (int i = 1; i < TOK; ++i) {
            int key = chosen[i], j = i - 1;
            while (j >= 0 && chosen[j] > key) { chosen[j + 1] = chosen[j]; --j; }
            chosen[j + 1] = key;
        }
        for (int t = 0; t < TOK; ++t) index[b * TOK + t] = chosen[t];
    }
}

// ---------------------------------------------------------------------------
// K6: out[b,t,n,c] = x[b, idx[b,t], n, c] + pos[t, c]   (float4 vectorized)
// ---------------------------------------------------------------------------
__global__ void k_gather(const float* __restrict__ x,
                         const float* __restrict__ pos,
                         const int* __restrict__ index,
                         float* __restrict__ out) {
    const long long total4 = (long long)B_ * TOK * N_ * (C_ / 4);
    long long e = (long long)blockIdx.x * blockDim.x + threadIdx.x;
    if (e >= total4) return;
    const int c4 = (int)(e & (C_ / 4 - 1));
    long long r  = e >> 7;                   // C_/4 == 128
    const int n  = (int)(r % N_); r /= N_;
    const int t  = (int)(r % TOK);
    const int b  = (int)(r / TOK);
    const int f  = index[b * TOK + t];

    float4 xv = ((const float4*)x)[(((size_t)b * F_ + f) * N_ + n) * (C_ / 4) + c4];
    float4 pv = ((const float4*)pos)[(size_t)t * (C_ / 4) + c4];
    xv.x += pv.x; xv.y += pv.y; xv.z += pv.z; xv.w += pv.w;
    ((float4*)out)[e] = xv;
}

// ---------------------------------------------------------------------------
extern "C" void kernel_launch(void* const* d_in, const int* in_sizes, int n_in,
                              void* d_out, int out_size, void* d_ws, size_t ws_size,
                              hipStream_t stream) {
    const float* x     = (const float*)d_in[0];   // (8,1024,17,512)
    const float* pos   = (const float*)d_in[1];   // (1,81,512)
    const float* noise = (const float*)d_in[2];   // (8,1024)
    // d_in[3] = current_layer (always LAYER_INDEX in harness setup)

    // workspace layout
    float*    xknn  = (float*)d_ws;                         // B*F*C
    float*    sq    = xknn + (size_t)B_ * F_ * C_;          // B*F
    float*    dist  = sq + (size_t)B_ * F_;                 // B*F*F
    float*    dens  = dist + (size_t)B_ * F_ * F_;          // B*F
    float*    scor  = dens + (size_t)B_ * F_;               // B*F
    unsigned* dmax  = (unsigned*)(scor + (size_t)B_ * F_);  // B
    int*      index = (int*)(dmax + B_);                    // B*TOK
    float*    out   = (float*)d_out;

    k_meanpool<<<B_ * F_, 128, 0, stream>>>(x, xknn, sq, dmax);

    const int waves = B_ * FTB * FTB;                       // 2048 waves
    k_gram_dist<<<waves / 4, 128, 0, stream>>>(xknn, sq, dist, dmax);

    k_density<<<B_ * F_, 256, 0, stream>>>(dist, noise, dens);
    k_score  <<<B_ * F_, 256, 0, stream>>>(dist, dens, dmax, scor);
    k_topk   <<<B_,      256, 0, stream>>>(scor, index);

    const long long total4 = (long long)B_ * TOK * N_ * (C_ / 4);
    k_gather<<<(int)((total4 + 255) / 256), 256, 0, stream>>>(x, pos, index, out);
}